// LearnableJPEG_85624468013289
// MI455X (gfx1250) — compile-verified
//
#include <hip/hip_runtime.h>
#include <math.h>

typedef __attribute__((ext_vector_type(2))) float v2f;
typedef __attribute__((ext_vector_type(8))) float v8f;

#define LDSW   20            // LDS row stride (dwords): conflict-free rows+cols, 16B-aligned rows
#define IMG_W  256
#define IMG_HW (256*256)

// One wave = one 16x16 pixel tile (2x2 JPEG blocks), all 3 channels.
// DCT of tile = A16 * X * A16^T with A16 = blockdiag(D8, D8); IDCT = A16^T * C * A16.
// Each 16x16x16 matmul = 4 chained V_WMMA_F32_16X16X4_F32 (exact fp32).
__global__ __launch_bounds__(256) void jpeg_tile_kernel(
    const float* __restrict__ x, const float* __restrict__ qlum,
    const float* __restrict__ qchr, float* __restrict__ out, int nTiles)
{
  __shared__ __align__(16) float smem[8][5][16 * LDSW];   // per-wave: Y,Cb,Cr,TMP,DCT
  const int lane = threadIdx.x & 31;
  const int wv   = threadIdx.x >> 5;
  const int tile = blockIdx.x * 8 + wv;
  if (tile >= nTiles) return;

  const int img = tile >> 8;           // 256 tiles per 256x256 image
  const int th  = (tile >> 4) & 15;
  const int tw  = tile & 15;
  const int h0  = th * 16, w0 = tw * 16;
  const size_t base = (size_t)img * 3 * IMG_HW;

  float* CH0 = smem[wv][0];
  float* CH1 = smem[wv][1];
  float* CH2 = smem[wv][2];
  float* TMP = smem[wv][3];
  float* DCT = smem[wv][4];
  float* CH[3] = { CH0, CH1, CH2 };

  const int lh  = lane >> 4;    // which 16-lane half
  const int l16 = lane & 15;

  // ---- Build A16 = blockdiag(D,D), D = 8x8 DCT-II matrix (once per wave) ----
#pragma unroll
  for (int i = 0; i < 8; ++i) {
    int idx = lane + 32 * i;            // 0..255
    int m = idx >> 4, k = idx & 15;
    float v = 0.0f;
    if ((m >> 3) == (k >> 3)) {         // diagonal 8x8 blocks only
      int u = m & 7, n = k & 7;
      v = (u == 0) ? 0.35355339059f     // sqrt(1/8)
                   : 0.5f * cosf(0.196349540849362f * (float)(u * (2 * n + 1)));
    }
    DCT[m * LDSW + k] = v;
  }

  // ---- Load 16x16 RGB tile (float4 coalesced), convert to YCbCr into LDS ----
  const float* Xr = x + base;
  const float* Xg = Xr + IMG_HW;
  const float* Xb = Xg + IMG_HW;
#pragma unroll
  for (int i = 0; i < 2; ++i) {
    int row = i * 8 + (lane >> 2);
    int col = (lane & 3) * 4;
    size_t off = (size_t)(h0 + row) * IMG_W + (w0 + col);
    float4 r4 = *(const float4*)(Xr + off);
    float4 g4 = *(const float4*)(Xg + off);
    float4 b4 = *(const float4*)(Xb + off);
    float4 y4, cb4, cr4;
#define CCONV(c)                                                                 \
    y4.c  =  0.299f    * r4.c + 0.587f    * g4.c + 0.114f    * b4.c;             \
    cb4.c = -0.168736f * r4.c - 0.331264f * g4.c + 0.5f      * b4.c + 128.0f;    \
    cr4.c =  0.5f      * r4.c - 0.418688f * g4.c - 0.081312f * b4.c + 128.0f;
    CCONV(x) CCONV(y) CCONV(z) CCONV(w)
#undef CCONV
    *(float4*)(CH0 + row * LDSW + col) = y4;
    *(float4*)(CH1 + row * LDSW + col) = cb4;
    *(float4*)(CH2 + row * LDSW + col) = cr4;
  }

  // ---- WMMA operand fetch patterns (ISA 16x16x4 f32 layouts) ----
  // ldP: element [l16][4j+v+2*lh] of a plain row-major 16x16 LDS array
  // ldQ: element [4j+v+2*lh][l16]
  auto ldP = [&](const float* buf, int j, int v) -> float {
    return buf[l16 * LDSW + 4 * j + v + 2 * lh];
  };
  auto ldQ = [&](const float* buf, int j, int v) -> float {
    return buf[(4 * j + v + 2 * lh) * LDSW + l16];
  };
  // 16x16x16 matmul via 4 chained K=4 WMMAs. Ar/Br pick the fetch pattern so
  // the same helper serves data-as-A, data-as-B, const, and transposed-const.
  auto MM = [&](const float* Ab, int Ar, const float* Bb, int Br) -> v8f {
    v8f acc = {0.f, 0.f, 0.f, 0.f, 0.f, 0.f, 0.f, 0.f};
#pragma unroll
    for (int j = 0; j < 4; ++j) {
      v2f a, b;
      a.x = Ar ? ldP(Ab, j, 0) : ldQ(Ab, j, 0);
      a.y = Ar ? ldP(Ab, j, 1) : ldQ(Ab, j, 1);
      b.x = Br ? ldP(Bb, j, 0) : ldQ(Bb, j, 0);
      b.y = Br ? ldP(Bb, j, 1) : ldQ(Bb, j, 1);
      acc = __builtin_amdgcn_wmma_f32_16x16x4_f32(false, a, false, b,
                                                  (short)0, acc, false, false);
    }
    return acc;
  };
  // Store C/D-layout accumulator as plain row-major 16x16 in LDS.
  auto ST = [&](float* buf, v8f v) {
#pragma unroll
    for (int r = 0; r < 8; ++r) buf[(r + 8 * lh) * LDSW + l16] = v[r];
  };

  // ---- Per-channel: DCT -> quant/dequant -> IDCT (all WMMA + LDS restage) ----
  v8f rec[3];
#pragma unroll
  for (int c = 0; c < 3; ++c) {
    const float* qt = (c == 0) ? qlum : qchr;

    v8f u1 = MM(DCT, 1, CH[c], 0);      // A16 * X
    ST(TMP, u1);
    v8f coef = MM(TMP, 1, DCT, 1);      // (A16 X) * A16^T  -> DCT coefficients

    // quant/dequant: C/D layout => qu = r, qv = lane&7 (blockdiag => M%8, N%8)
    // qs in [1,2]; v_rcp_f32 (~1 ulp) instead of IEEE div: coeff error ~5e-4,
    // far below the unit quant step -> numerically safe, saves ~10 VALU/div.
#pragma unroll
    for (int r = 0; r < 8; ++r) {
      float qs = qt[r * 8 + (lane & 7)] * 1.0e5f;   // q / S, S = 1e-5
      float rq = __builtin_amdgcn_rcpf(qs);
      coef[r] = rintf(coef[r] * rq) * qs;           // round-half-even == jnp.round
    }
    ST(TMP, coef);

    v8f w1 = MM(DCT, 0, TMP, 0);        // A16^T * C
    ST(CH[c], w1);
    rec[c] = MM(CH[c], 1, DCT, 0);      // (A16^T C) * A16  -> reconstructed channel
  }

  // ---- YCbCr -> RGB, clamp, restage to LDS for coalesced float4 stores ----
#pragma unroll
  for (int r = 0; r < 8; ++r) {
    float y  = rec[0][r];
    float cb = rec[1][r] - 128.0f;
    float cr = rec[2][r] - 128.0f;
    float R = y + 1.402f * cr;
    float G = y - 0.344136f * cb - 0.714136f * cr;
    float Bv = y + 1.722f * cb;
    R  = fminf(fmaxf(R,  0.0f), 255.0f);
    G  = fminf(fmaxf(G,  0.0f), 255.0f);
    Bv = fminf(fmaxf(Bv, 0.0f), 255.0f);
    int m = r + 8 * lh;
    CH0[m * LDSW + l16] = R;
    CH1[m * LDSW + l16] = G;
    CH2[m * LDSW + l16] = Bv;
  }

  float* Or = out + base;
  float* Og = Or + IMG_HW;
  float* Ob = Og + IMG_HW;
#pragma unroll
  for (int i = 0; i < 2; ++i) {
    int row = i * 8 + (lane >> 2);
    int col = (lane & 3) * 4;
    size_t off = (size_t)(h0 + row) * IMG_W + (w0 + col);
    *(float4*)(Or + off) = *(const float4*)(CH0 + row * LDSW + col);
    *(float4*)(Og + off) = *(const float4*)(CH1 + row * LDSW + col);
    *(float4*)(Ob + off) = *(const float4*)(CH2 + row * LDSW + col);
  }
}

extern "C" void kernel_launch(void* const* d_in, const int* in_sizes, int n_in,
                              void* d_out, int out_size, void* d_ws, size_t ws_size,
                              hipStream_t stream) {
  const float* x  = (const float*)d_in[0];
  const float* ql = (const float*)d_in[1];
  const float* qc = (const float*)d_in[2];
  float* o = (float*)d_out;
  (void)d_ws; (void)ws_size; (void)n_in; (void)out_size;

  int nImg   = in_sizes[0] / (3 * IMG_HW);   // 128
  int nTiles = nImg * 256;                   // 16x16 tiles per image
  int blocks = (nTiles + 7) / 8;             // 8 waves (tiles) per 256-thread block
  jpeg_tile_kernel<<<blocks, 256, 0, stream>>>(x, ql, qc, o, nTiles);
}